// StructuralAttention_51041391346250
// MI455X (gfx1250) — compile-verified
//
#include <hip/hip_runtime.h>

#define N_NODES 100000
#define D 64
#define N_EDGES 1280000
#define LN_EPS 1e-5f

typedef __attribute__((ext_vector_type(2))) float v2f;
typedef __attribute__((ext_vector_type(8))) float v8f;

// ---------------------------------------------------------------------------
// Pass 1: qk = x @ qk_w^T + qk_b ; v = x @ v_w^T + v_b
// One wave computes a 16x64 tile of both outputs with fp32 WMMA (16x16x4).
// A fragment (whole K=64) preloaded into 16 v2f regs; B loads batched per
// K-step so the compiler emits load-clauses instead of per-WMMA waits.
// Also zero-initializes unnorm/seg for the edge pass (replaces memsets).
// ---------------------------------------------------------------------------
__global__ __launch_bounds__(128)
void proj_qk_v(const float* __restrict__ x,
               const float* __restrict__ qk_w, const float* __restrict__ qk_b,
               const float* __restrict__ v_w,  const float* __restrict__ v_b,
               float* __restrict__ qk, float* __restrict__ v,
               float* __restrict__ unnorm, float* __restrict__ seg)
{
    const int lane = threadIdx.x & 31;
    const int wave = threadIdx.x >> 5;
    const int row0 = (blockIdx.x * 4 + wave) * 16;
    if (row0 >= N_NODES) return;   // wave-uniform: EXEC all-ones for WMMA

    const int mh   = lane & 15;        // A-row / B-col within tile
    const int koff = (lane >> 4) * 2;  // K sub-offset for this lane half

    // ---- preload full A fragment: lane holds A[mh][koff + 4i + {0,1}] ----
    const int arow = min(row0 + mh, N_NODES - 1);
    const float* aptr = x + (size_t)arow * D + koff;
    v2f a[16];
    #pragma unroll
    for (int i = 0; i < 16; ++i) a[i] = *(const v2f*)(aptr + i * 4);

    const v8f zero8 = {0.f,0.f,0.f,0.f,0.f,0.f,0.f,0.f};
    v8f cq[4] = {zero8, zero8, zero8, zero8};
    v8f cv[4] = {zero8, zero8, zero8, zero8};

    const float* wq = qk_w + mh * D + koff;   // row mh (+16*nb) of weights
    const float* wv = v_w  + mh * D + koff;

    #pragma unroll
    for (int i = 0; i < 16; ++i) {
        const int kb = i * 4;
        v2f bq[4], bv[4];
        #pragma unroll
        for (int nb = 0; nb < 4; ++nb) {
            bq[nb] = *(const v2f*)(wq + nb * 16 * D + kb);
            bv[nb] = *(const v2f*)(wv + nb * 16 * D + kb);
        }
        #pragma unroll
        for (int nb = 0; nb < 4; ++nb) {
            cq[nb] = __builtin_amdgcn_wmma_f32_16x16x4_f32(
                false, a[i], false, bq[nb], (short)0, cq[nb], false, false);
            cv[nb] = __builtin_amdgcn_wmma_f32_16x16x4_f32(
                false, a[i], false, bv[nb], (short)0, cv[nb], false, false);
        }
    }

    // ---- store + zero-init scratch for the edge pass ----
    if (lane < 16 && row0 + lane < N_NODES) seg[row0 + lane] = 0.f;
    #pragma unroll
    for (int nb = 0; nb < 4; ++nb) {
        const int col = nb * 16 + mh;
        const float bq = qk_b[col], bv = v_b[col];
        #pragma unroll
        for (int r = 0; r < 8; ++r) {
            const int row = row0 + r + 8 * (lane >> 4);
            if (row < N_NODES) {
                qk    [(size_t)row * D + col] = cq[nb][r] + bq;
                v     [(size_t)row * D + col] = cv[nb][r] + bv;
                unnorm[(size_t)row * D + col] = 0.f;
            }
        }
    }
}

// ---------------------------------------------------------------------------
// Pass 2 (single edge pass): per edge
//   s = clamp(<qk[dst], qk[src]>/8, -5, 5);  w = exp(s)
//   seg[dst] += w ;  unnorm[dst,:] += w * v[src,:]   (fp32 HW atomics)
// Max-subtraction unnecessary: clamp bounds exp(s) to [6.7e-3, 148];
// normalization deferred to pass 3 (out = unnorm/seg).
// ---------------------------------------------------------------------------
__global__ __launch_bounds__(256)
void edge_pass(const long long* __restrict__ ei,
               const float* __restrict__ qk, const float* __restrict__ v,
               float* __restrict__ unnorm, float* __restrict__ seg)
{
    const int e = (blockIdx.x * blockDim.x + threadIdx.x) >> 4;
    if (e >= N_EDGES) return;
    const int sub = threadIdx.x & 15;

    const int src = (int)ei[e];
    const int dst = (int)ei[N_EDGES + e];

    const float4 qd = *(const float4*)(qk + (size_t)dst * D + sub * 4);
    const float4 qs = *(const float4*)(qk + (size_t)src * D + sub * 4);
    float p = qd.x*qs.x + qd.y*qs.y + qd.z*qs.z + qd.w*qs.w;
    #pragma unroll
    for (int off = 8; off >= 1; off >>= 1)
        p += __shfl_xor(p, off, 16);          // butterfly: all lanes get total

    float s = p * 0.125f;                      // 1/sqrt(64)
    s = fminf(5.f, fmaxf(-5.f, s));
    const float w = __expf(s);

    if (sub == 0) unsafeAtomicAdd(&seg[dst], w);

    const float4 vs = *(const float4*)(v + (size_t)src * D + sub * 4);
    float* o = unnorm + (size_t)dst * D + sub * 4;
    unsafeAtomicAdd(o + 0, w * vs.x);
    unsafeAtomicAdd(o + 1, w * vs.y);
    unsafeAtomicAdd(o + 2, w * vs.z);
    unsafeAtomicAdd(o + 3, w * vs.w);
}

// ---------------------------------------------------------------------------
// Pass 3: h = (unnorm / seg) @ o_w^T + o_b + x   (fp32 WMMA, same tiling)
// ---------------------------------------------------------------------------
__global__ __launch_bounds__(128)
void out_proj(const float* __restrict__ unnorm, const float* __restrict__ seg,
              const float* __restrict__ o_w,   const float* __restrict__ o_b,
              const float* __restrict__ x,     float* __restrict__ h)
{
    const int lane = threadIdx.x & 31;
    const int wave = threadIdx.x >> 5;
    const int row0 = (blockIdx.x * 4 + wave) * 16;
    if (row0 >= N_NODES) return;

    const int mh   = lane & 15;
    const int koff = (lane >> 4) * 2;

    const int arow = min(row0 + mh, N_NODES - 1);
    const float ssum = seg[arow];
    const float inv  = ssum > 0.f ? 1.f / ssum : 0.f;   // empty segment -> 0

    const float* aptr = unnorm + (size_t)arow * D + koff;
    v2f a[16];
    #pragma unroll
    for (int i = 0; i < 16; ++i) {
        a[i] = *(const v2f*)(aptr + i * 4);
        a[i].x *= inv; a[i].y *= inv;
    }

    const v8f zero8 = {0.f,0.f,0.f,0.f,0.f,0.f,0.f,0.f};
    v8f c[4] = {zero8, zero8, zero8, zero8};

    const float* wo = o_w + mh * D + koff;
    #pragma unroll
    for (int i = 0; i < 16; ++i) {
        const int kb = i * 4;
        v2f b[4];
        #pragma unroll
        for (int nb = 0; nb < 4; ++nb)
            b[nb] = *(const v2f*)(wo + nb * 16 * D + kb);
        #pragma unroll
        for (int nb = 0; nb < 4; ++nb)
            c[nb] = __builtin_amdgcn_wmma_f32_16x16x4_f32(
                false, a[i], false, b[nb], (short)0, c[nb], false, false);
    }

    #pragma unroll
    for (int nb = 0; nb < 4; ++nb) {
        const int col = nb * 16 + mh;
        const float bo = o_b[col];
        #pragma unroll
        for (int r = 0; r < 8; ++r) {
            const int row = row0 + r + 8 * (lane >> 4);
            if (row < N_NODES)
                h[(size_t)row * D + col] = c[nb][r] + bo + x[(size_t)row * D + col];
        }
    }
}

// ---------------------------------------------------------------------------
// Pass 4: LayerNorm over D=64 per row; one wave per row, 2 elements per lane.
// ---------------------------------------------------------------------------
__global__ __launch_bounds__(256)
void layernorm_k(const float* __restrict__ h, const float* __restrict__ g,
                 const float* __restrict__ b, float* __restrict__ out)
{
    const int row = blockIdx.x * 8 + (threadIdx.x >> 5);
    if (row >= N_NODES) return;
    const int lane = threadIdx.x & 31;

    const float a0 = h[(size_t)row * D + lane];
    const float a1 = h[(size_t)row * D + 32 + lane];

    float sum = a0 + a1;
    #pragma unroll
    for (int off = 16; off >= 1; off >>= 1) sum += __shfl_xor(sum, off, 32);
    const float mu = sum * (1.f / 64.f);

    const float d0 = a0 - mu, d1 = a1 - mu;
    float vs = d0 * d0 + d1 * d1;
    #pragma unroll
    for (int off = 16; off >= 1; off >>= 1) vs += __shfl_xor(vs, off, 32);
    const float rstd = rsqrtf(vs * (1.f / 64.f) + LN_EPS);

    out[(size_t)row * D + lane]      = d0 * rstd * g[lane]      + b[lane];
    out[(size_t)row * D + 32 + lane] = d1 * rstd * g[32 + lane] + b[32 + lane];
}

// ---------------------------------------------------------------------------
extern "C" void kernel_launch(void* const* d_in, const int* in_sizes, int n_in,
                              void* d_out, int out_size, void* d_ws, size_t ws_size,
                              hipStream_t stream)
{
    const float*     x    = (const float*)d_in[0];
    const long long* ei   = (const long long*)d_in[1];   // int64 [2, E]
    const float*     qk_w = (const float*)d_in[2];
    const float*     qk_b = (const float*)d_in[3];
    const float*     v_w  = (const float*)d_in[4];
    const float*     v_b  = (const float*)d_in[5];
    const float*     o_w  = (const float*)d_in[6];
    const float*     o_b  = (const float*)d_in[7];
    const float*     ln_g = (const float*)d_in[8];
    const float*     ln_b = (const float*)d_in[9];

    // workspace layout (floats): qk | v | unnorm | seg    (~77.2 MB)
    float* qk  = (float*)d_ws;
    float* v   = qk  + (size_t)N_NODES * D;
    float* un  = v   + (size_t)N_NODES * D;
    float* seg = un  + (size_t)N_NODES * D;
    float* h   = qk;   // qk dead after edge_pass -> reuse for residual sum

    const int tiles   = (N_NODES + 15) / 16;   // 6250 16-row tiles
    const int pblocks = (tiles + 3) / 4;       // 4 waves per block

    // pass 1 also zero-inits un/seg (replaces memsets)
    proj_qk_v<<<pblocks, 128, 0, stream>>>(x, qk_w, qk_b, v_w, v_b, qk, v, un, seg);

    const int eblocks = (N_EDGES * 16 + 255) / 256;   // 16 lanes per edge
    edge_pass<<<eblocks, 256, 0, stream>>>(ei, qk, v, un, seg);

    out_proj<<<pblocks, 128, 0, stream>>>(un, seg, o_w, o_b, x, h);

    layernorm_k<<<(N_NODES + 7) / 8, 256, 0, stream>>>(h, ln_g, ln_b, (float*)d_out);
}